// param_PALM_1941325218459
// MI455X (gfx1250) — compile-verified
//
#include <hip/hip_runtime.h>
#include <math.h>

// Problem constants (match reference)
#define Bb 32
#define Mm 256
#define Kd 512   // n_atoms
#define Nn 512
#define Tt 10

// Tiling
#define KT  32          // reduce-dim chunk staged in LDS
#define LDA 36          // padded LDS row stride for A-operand tile (128 x 32)
#define LDB 68          // padded LDS row stride for B-operand tile (32 x 64)

typedef float v2f __attribute__((ext_vector_type(2)));
typedef float v8f __attribute__((ext_vector_type(8)));

// ---------------------------------------------------------------------------
// CDNA5 async global->LDS path (GLOBAL_LOAD_ASYNC_TO_LDS_*, ASYNCcnt).
// Builtin prototypes (from clang diagnostics):
//   b128: (global v4i*, local v4i*, Ii offset, Ii cpol)
//   b32 : (global int*,  local int*,  Ii offset, Ii cpol)
// ---------------------------------------------------------------------------
#if defined(__has_builtin)
#if __has_builtin(__builtin_amdgcn_global_load_async_to_lds_b128) && \
    __has_builtin(__builtin_amdgcn_global_load_async_to_lds_b32) &&  \
    __has_builtin(__builtin_amdgcn_s_wait_asynccnt)
#define HAVE_ASYNC 1
#endif
#endif
#ifndef HAVE_ASYNC
#define HAVE_ASYNC 0
#endif

#if HAVE_ASYNC
typedef int i32x4_vs __attribute__((vector_size(16)));
typedef __attribute__((address_space(1))) i32x4_vs* gp_v4i;
typedef __attribute__((address_space(3))) i32x4_vs* lp_v4i;
typedef __attribute__((address_space(1))) int*      gp_i32;
typedef __attribute__((address_space(3))) int*      lp_i32;
#endif

__device__ __forceinline__ void cp_g2l_b128(const float* g, float* l) {
#if HAVE_ASYNC
  __builtin_amdgcn_global_load_async_to_lds_b128((gp_v4i)(g), (lp_v4i)(l), 0, 0);
#else
  *(float4*)l = *(const float4*)g;
#endif
}

__device__ __forceinline__ void cp_g2l_b32(const float* g, float* l) {
#if HAVE_ASYNC
  __builtin_amdgcn_global_load_async_to_lds_b32((gp_i32)(g), (lp_i32)(l), 0, 0);
#else
  *l = *g;
#endif
}

template <int N>
__device__ __forceinline__ void wait_async() {
#if HAVE_ASYNC
  __builtin_amdgcn_s_wait_asynccnt(N);
#endif
}

// ---------------------------------------------------------------------------
// Wave-level 32x32 fp32 WMMA micro-kernel over one staged KT slice.
// ldsA: [128 rows][LDA]   row = output-row dim, inner = reduce dim
// ldsB: [KT rows][LDB]    row = reduce dim,     inner = output-col dim
// Fragment layouts per CDNA5 ISA 7.12.2 (16x4 A, 4x16 B, 16x16 C/D).
// ---------------------------------------------------------------------------
__device__ __forceinline__ void wave_mma_32x32(const float* __restrict__ ldsA,
                                               const float* __restrict__ ldsB,
                                               int wr, int wc, int lane,
                                               v8f acc[2][2]) {
  const int half = lane >> 4;       // 0: lanes 0-15, 1: lanes 16-31
  const int lm   = lane & 15;
  const int r0 = wr * 32 + lm;
  const int c0 = wc * 32 + lm;
#pragma unroll
  for (int kk = 0; kk < KT; kk += 4) {
    const int k = kk + 2 * half;
    v2f a0, a1, b0, b1;
    a0.x = ldsA[r0 * LDA + k];            a0.y = ldsA[r0 * LDA + k + 1];
    a1.x = ldsA[(r0 + 16) * LDA + k];     a1.y = ldsA[(r0 + 16) * LDA + k + 1];
    b0.x = ldsB[k * LDB + c0];            b0.y = ldsB[(k + 1) * LDB + c0];
    b1.x = ldsB[k * LDB + c0 + 16];       b1.y = ldsB[(k + 1) * LDB + c0 + 16];
    acc[0][0] = __builtin_amdgcn_wmma_f32_16x16x4_f32(false, a0, false, b0, (short)0, acc[0][0], false, false);
    acc[0][1] = __builtin_amdgcn_wmma_f32_16x16x4_f32(false, a0, false, b1, (short)0, acc[0][1], false, false);
    acc[1][0] = __builtin_amdgcn_wmma_f32_16x16x4_f32(false, a1, false, b0, (short)0, acc[1][0], false, false);
    acc[1][1] = __builtin_amdgcn_wmma_f32_16x16x4_f32(false, a1, false, b1, (short)0, acc[1][1], false, false);
  }
}

// ---------------------------------------------------------------------------
// resid[b] = A[b](MxK) * S[b](KxN) - X[b]   ; grid (N/64, M/128, B), 256 thr
// AINIT: A == 1/16 everywhere (step 0)
// ---------------------------------------------------------------------------
template <int AINIT>
__global__ void k_resid(const float* __restrict__ A, const float* __restrict__ S,
                        const float* __restrict__ X, float* __restrict__ resid) {
  __shared__ __align__(16) float ldsA[2][128 * LDA];
  __shared__ __align__(16) float ldsB[2][KT * LDB];
  const int b  = blockIdx.z;
  const int m0 = blockIdx.y * 128;
  const int n0 = blockIdx.x * 64;
  const int tid  = threadIdx.x;
  const int lane = tid & 31;
  const int wave = tid >> 5;
  const int wr = wave >> 1, wc = wave & 1;

  const float* Ab = A + (size_t)b * Mm * Kd;
  const float* Sb = S + (size_t)b * Kd * Nn;
  v8f acc[2][2] = {};

  if (AINIT) {  // constant A tile: fill both buffers once
    for (int i = tid; i < 128 * LDA; i += 256) {
      ldsA[0][i] = 0.0625f; ldsA[1][i] = 0.0625f;
    }
  }

  // stage issue: async copies for slice kc into buffer s
  auto issue = [&](int s, int kc) {
    if (!AINIT) {
#pragma unroll
      for (int i = 0; i < 4; ++i) {     // A tile 128x32: 4 b128 per wave
        int idx = tid + i * 256;
        int row = idx >> 3;
        int c4  = (idx & 7) * 4;
        cp_g2l_b128(Ab + (size_t)(m0 + row) * Kd + kc + c4, &ldsA[s][row * LDA + c4]);
      }
    }
#pragma unroll
    for (int i = 0; i < 2; ++i) {       // B tile 32x64: 2 b128 per wave
      int idx = tid + i * 256;
      int row = idx >> 4;
      int c4  = (idx & 15) * 4;
      cp_g2l_b128(Sb + (size_t)(kc + row) * Nn + n0 + c4, &ldsB[s][row * LDB + c4]);
    }
  };
  constexpr int NAS = AINIT ? 2 : 6;    // async ops issued per wave per stage

  issue(0, 0);
  for (int kc = 0; kc < Kd; kc += KT) {
    const int s = (kc >> 5) & 1;
    if (kc + KT < Kd) { issue(s ^ 1, kc + KT); wait_async<NAS>(); }
    else              { wait_async<0>(); }
    __syncthreads();
    wave_mma_32x32(ldsA[s], ldsB[s], wr, wc, lane, acc);
    __syncthreads();
  }

  const float* Xb = X + (size_t)b * Mm * Nn;
  float* Rb = resid + (size_t)b * Mm * Nn;
  const int half = lane >> 4, lm = lane & 15;
#pragma unroll
  for (int rt = 0; rt < 2; ++rt)
#pragma unroll
    for (int ct = 0; ct < 2; ++ct)
#pragma unroll
      for (int r = 0; r < 8; ++r) {
        int m = m0 + wr * 32 + rt * 16 + half * 8 + r;
        int n = n0 + wc * 32 + ct * 16 + lm;
        Rb[(size_t)m * Nn + n] = acc[rt][ct][r] - Xb[(size_t)m * Nn + n];
      }
}

// resid = -X (step 0, S == 0)
__global__ void k_resid_init(const float* __restrict__ X, float* __restrict__ resid, int n) {
  int i = blockIdx.x * blockDim.x + threadIdx.x;
  if (i < n) resid[i] = -X[i];
}

// ---------------------------------------------------------------------------
// Snew = prox( Sprev - (1/L) * A^T * resid, theta )
// grid (N/64, K/128, B), 256 thr. Reduce over M.
// ---------------------------------------------------------------------------
template <int AINIT, int SZERO>
__global__ void k_supdate(const float* __restrict__ A, const float* __restrict__ resid,
                          const float* __restrict__ Sprev,
                          const float* __restrict__ theta_t, const float* __restrict__ LS_t,
                          float* __restrict__ Snew) {
  __shared__ __align__(16) float ldsA[2][128 * LDA];   // A^T tile: [k 128][m 32]
  __shared__ __align__(16) float ldsB[2][KT * LDB];    // resid tile: [m 32][n 64]
  const int b  = blockIdx.z;
  const int k0 = blockIdx.y * 128;
  const int n0 = blockIdx.x * 64;
  const int tid  = threadIdx.x;
  const int lane = tid & 31;
  const int wave = tid >> 5;
  const int wr = wave >> 1, wc = wave & 1;

  const float th   = theta_t[0];
  const float invL = 1.0f / LS_t[0];
  const float* Ab = A + (size_t)b * Mm * Kd;
  const float* Rb = resid + (size_t)b * Mm * Nn;
  v8f acc[2][2] = {};

  if (AINIT) {
    for (int i = tid; i < 128 * LDA; i += 256) {
      ldsA[0][i] = 0.0625f; ldsA[1][i] = 0.0625f;
    }
  }

  auto issue = [&](int s, int mc) {
    if (!AINIT) {
      // A^T tile via per-lane b32 scatter: ldsA[k][m] = A[mc+m][k0+k]
#pragma unroll
      for (int i = 0; i < 16; ++i) {    // 4096 elements -> 16 b32 per wave
        int e    = tid + i * 256;
        int mrow = e >> 7;              // 0..31
        int kcol = e & 127;             // 0..127 (lanes sweep k: coalesced)
        cp_g2l_b32(Ab + (size_t)(mc + mrow) * Kd + k0 + kcol,
                   &ldsA[s][kcol * LDA + mrow]);
      }
    }
#pragma unroll
    for (int i = 0; i < 2; ++i) {       // resid tile 32x64: 2 b128 per wave
      int idx = tid + i * 256;
      int row = idx >> 4;
      int c4  = (idx & 15) * 4;
      cp_g2l_b128(Rb + (size_t)(mc + row) * Nn + n0 + c4, &ldsB[s][row * LDB + c4]);
    }
  };
  constexpr int NAS = AINIT ? 2 : 18;

  issue(0, 0);
  for (int mc = 0; mc < Mm; mc += KT) {
    const int s = (mc >> 5) & 1;
    if (mc + KT < Mm) { issue(s ^ 1, mc + KT); wait_async<NAS>(); }
    else              { wait_async<0>(); }
    __syncthreads();
    wave_mma_32x32(ldsA[s], ldsB[s], wr, wc, lane, acc);
    __syncthreads();
  }

  float* Sb = Snew + (size_t)b * Kd * Nn;
  const int half = lane >> 4, lm = lane & 15;
#pragma unroll
  for (int rt = 0; rt < 2; ++rt)
#pragma unroll
    for (int ct = 0; ct < 2; ++ct)
#pragma unroll
      for (int r = 0; r < 8; ++r) {
        int k = k0 + wr * 32 + rt * 16 + half * 8 + r;
        int n = n0 + wc * 32 + ct * 16 + lm;
        float sp = 0.0f;
        if (!SZERO) sp = Sprev[(size_t)b * Kd * Nn + (size_t)k * Nn + n];
        float v  = sp - acc[rt][ct][r] * invL;
        float av = fabsf(v) - th;
        Sb[(size_t)k * Nn + n] = (av > 0.0f) ? copysignf(av, v) : 0.0f;
      }
}

// ---------------------------------------------------------------------------
// Power iteration: sigma_max^2 of S[0] (512x512); writes invLA = 1/(1.001*s^2).
// Also zeroes colnorm2 [B*K]. Single workgroup, 512 threads.
// ---------------------------------------------------------------------------
__global__ void k_spectral(const float* __restrict__ S0, float* __restrict__ invLA,
                           float* __restrict__ colnorm2, int colnElems) {
  __shared__ float v[Nn];
  __shared__ float u[Kd];
  __shared__ float red[512];
  const int tid = threadIdx.x;

  for (int i = tid; i < colnElems; i += 512) colnorm2[i] = 0.0f;
  v[tid] = 1.0f;
  __syncthreads();

  for (int it = 0; it < 8; ++it) {
    float s = 0.0f;
    const float* row = S0 + (size_t)tid * Nn;
#pragma unroll 4
    for (int j = 0; j < Nn; j += 4) {
      float4 x = *(const float4*)(row + j);
      s += x.x * v[j] + x.y * v[j + 1] + x.z * v[j + 2] + x.w * v[j + 3];
    }
    u[tid] = s;
    __syncthreads();
    float w = 0.0f;
    for (int k = 0; k < Kd; ++k) w += S0[(size_t)k * Nn + tid] * u[k];
    red[tid] = w * w;
    __syncthreads();
    for (int off = 256; off > 0; off >>= 1) {
      if (tid < off) red[tid] += red[tid + off];
      __syncthreads();
    }
    float rn = rsqrtf(fmaxf(red[0], 1e-30f));
    __syncthreads();
    v[tid] = w * rn;
    __syncthreads();
  }

  float s = 0.0f;
  const float* row = S0 + (size_t)tid * Nn;
#pragma unroll 4
  for (int j = 0; j < Nn; j += 4) {
    float4 x = *(const float4*)(row + j);
    s += x.x * v[j] + x.y * v[j + 1] + x.z * v[j + 2] + x.w * v[j + 3];
  }
  red[tid] = s * s;
  __syncthreads();
  for (int off = 256; off > 0; off >>= 1) {
    if (tid < off) red[tid] += red[tid + off];
    __syncthreads();
  }
  if (tid == 0) invLA[0] = 1.0f / (1.001f * fmaxf(red[0], 1e-30f));
}

// ---------------------------------------------------------------------------
// Anew = Aprev - (resid2 * S^T) * invLA ; accumulate column sum-of-squares.
// grid (K/64, M/128, B), 256 thr. Reduce over N.
// ---------------------------------------------------------------------------
template <int AINIT>
__global__ void k_aupdate(const float* __restrict__ Aprev, const float* __restrict__ resid2,
                          const float* __restrict__ S, const float* __restrict__ invLAp,
                          float* __restrict__ Anew, float* __restrict__ colnorm2) {
  __shared__ __align__(16) float ldsA[2][128 * LDA];   // resid2 tile [m 128][n 32]
  __shared__ __align__(16) float ldsB[2][KT * LDB];    // S^T tile [n 32][k 64]
  __shared__ float colsq[64];
  const int b  = blockIdx.z;
  const int m0 = blockIdx.y * 128;
  const int k0 = blockIdx.x * 64;
  const int tid  = threadIdx.x;
  const int lane = tid & 31;
  const int wave = tid >> 5;
  const int wr = wave >> 1, wc = wave & 1;

  const float invLA = invLAp[0];
  const float* Rb = resid2 + (size_t)b * Mm * Nn;
  const float* Sb = S + (size_t)b * Kd * Nn;
  v8f acc[2][2] = {};

  auto issue = [&](int s, int nc) {
#pragma unroll
    for (int i = 0; i < 4; ++i) {       // resid2 tile 128x32: 4 b128 per wave
      int idx = tid + i * 256;
      int row = idx >> 3;
      int c4  = (idx & 7) * 4;
      cp_g2l_b128(Rb + (size_t)(m0 + row) * Nn + nc + c4, &ldsA[s][row * LDA + c4]);
    }
    // S^T tile via per-lane b32 scatter: ldsB[n][k] = S[k0+k][nc+n]
#pragma unroll
    for (int i = 0; i < 8; ++i) {       // 2048 elements -> 8 b32 per wave
      int e    = tid + i * 256;
      int krow = e >> 5;                // 0..63
      int ncol = e & 31;                // lanes sweep n: coalesced
      cp_g2l_b32(Sb + (size_t)(k0 + krow) * Nn + nc + ncol,
                 &ldsB[s][ncol * LDB + krow]);
    }
  };
  constexpr int NAS = 12;

  issue(0, 0);
  for (int nc = 0; nc < Nn; nc += KT) {
    const int s = (nc >> 5) & 1;
    if (nc + KT < Nn) { issue(s ^ 1, nc + KT); wait_async<NAS>(); }
    else              { wait_async<0>(); }
    __syncthreads();
    wave_mma_32x32(ldsA[s], ldsB[s], wr, wc, lane, acc);
    __syncthreads();
  }

  if (tid < 64) colsq[tid] = 0.0f;
  __syncthreads();

  float* Ab = Anew + (size_t)b * Mm * Kd;
  const int half = lane >> 4, lm = lane & 15;
#pragma unroll
  for (int rt = 0; rt < 2; ++rt)
#pragma unroll
    for (int ct = 0; ct < 2; ++ct) {
      int kc = k0 + wc * 32 + ct * 16 + lm;
      float ss = 0.0f;
#pragma unroll
      for (int r = 0; r < 8; ++r) {
        int m = m0 + wr * 32 + rt * 16 + half * 8 + r;
        float ap = AINIT ? 0.0625f
                         : Aprev[(size_t)b * Mm * Kd + (size_t)m * Kd + kc];
        float an = ap - acc[rt][ct][r] * invLA;
        Ab[(size_t)m * Kd + kc] = an;
        ss += an * an;
      }
      atomicAdd(&colsq[wc * 32 + ct * 16 + lm], ss);
    }
  __syncthreads();
  if (tid < 64) atomicAdd(&colnorm2[b * Kd + k0 + tid], colsq[tid]);
}

// Column renormalization: A[:,k] /= ||A[:,k]|| iff norm > 1
__global__ void k_renorm(float* __restrict__ A, const float* __restrict__ colnorm2, int n) {
  int i = blockIdx.x * blockDim.x + threadIdx.x;
  if (i >= n) return;
  int b = i / (Mm * Kd);
  int k = i % Kd;
  float n2 = colnorm2[b * Kd + k];
  if (n2 > 1.0f) A[i] *= (1.0f / sqrtf(n2));
}

// ---------------------------------------------------------------------------
extern "C" void kernel_launch(void* const* d_in, const int* in_sizes, int n_in,
                              void* d_out, int out_size, void* d_ws, size_t ws_size,
                              hipStream_t stream) {
  (void)in_sizes; (void)n_in; (void)out_size; (void)ws_size;
  const float* X     = (const float*)d_in[0];   // [B,M,N]
  const float* theta = (const float*)d_in[1];   // [T]
  const float* LS    = (const float*)d_in[2];   // [T]

  const size_t BKN = (size_t)Bb * Kd * Nn;
  const size_t BMK = (size_t)Bb * Mm * Kd;
  const size_t BMN = (size_t)Bb * Mm * Nn;

  float* out   = (float*)d_out;
  float* S_fin = out;
  float* A_fin = out + BKN;
  float* S_s   = A_fin + BMK;                   // [T,B,K,N]
  float* A_s   = S_s + (size_t)Tt * BKN;        // [T,B,M,K]

  float* resid    = (float*)d_ws;               // [B,M,N]
  float* colnorm2 = resid + BMN;                // [B,K]
  float* invLA    = colnorm2 + (size_t)Bb * Kd; // [1]

  const dim3 blk(256);
  const dim3 gResid(Nn / 64, Mm / 128, Bb);     // (8,2,32)
  const dim3 gSupd (Nn / 64, Kd / 128, Bb);     // (8,4,32)
  const dim3 gAupd (Kd / 64, Mm / 128, Bb);     // (8,2,32)

  for (int t = 0; t < Tt; ++t) {
    const int step0 = (t == 0);
    const float* Sprev = step0 ? nullptr : (S_s + (size_t)(t - 1) * BKN);
    const float* Aprev = step0 ? nullptr : (A_s + (size_t)(t - 1) * BMK);
    float* Scur = S_s + (size_t)t * BKN;
    float* Acur = A_s + (size_t)t * BMK;

    // 1) resid = A*S - X
    if (step0) {
      int n = (int)BMN;
      k_resid_init<<<dim3((n + 255) / 256), blk, 0, stream>>>(X, resid, n);
    } else {
      k_resid<0><<<gResid, blk, 0, stream>>>(Aprev, Sprev, X, resid);
    }
    // 2) S = prox(Sprev - (1/L) A^T resid)
    if (step0)
      k_supdate<1, 1><<<gSupd, blk, 0, stream>>>(Aprev, resid, Sprev, theta + t, LS + t, Scur);
    else
      k_supdate<0, 0><<<gSupd, blk, 0, stream>>>(Aprev, resid, Sprev, theta + t, LS + t, Scur);
    // 3) resid2 = A*Snew - X
    if (step0)
      k_resid<1><<<gResid, blk, 0, stream>>>(Aprev, Scur, X, resid);
    else
      k_resid<0><<<gResid, blk, 0, stream>>>(Aprev, Scur, X, resid);
    // 4) spectral norm of Snew[0] -> invLA; zero colnorm2
    k_spectral<<<dim3(1), dim3(512), 0, stream>>>(Scur, invLA, colnorm2, Bb * Kd);
    // 5) A = Aprev - (resid2 * S^T) * invLA ; accumulate column sumsq
    if (step0)
      k_aupdate<1><<<gAupd, blk, 0, stream>>>(Aprev, resid, Scur, invLA, Acur, colnorm2);
    else
      k_aupdate<0><<<gAupd, blk, 0, stream>>>(Aprev, resid, Scur, invLA, Acur, colnorm2);
    // 6) column renorm
    {
      int n = (int)BMK;
      k_renorm<<<dim3((n + 255) / 256), blk, 0, stream>>>(Acur, colnorm2, n);
    }
  }

  (void)hipMemcpyAsync(S_fin, S_s + (size_t)(Tt - 1) * BKN, BKN * sizeof(float),
                       hipMemcpyDeviceToDevice, stream);
  (void)hipMemcpyAsync(A_fin, A_s + (size_t)(Tt - 1) * BMK, BMK * sizeof(float),
                       hipMemcpyDeviceToDevice, stream);
}